// MinMax_33758442947087
// MI455X (gfx1250) — compile-verified
//
#include <hip/hip_runtime.h>
#include <stdint.h>

#define B_DIM 1024
#define I_DIM 512
#define O_DIM 512

#define BM 64
#define BN 64
#define BK 32
#define THREADS 256
#define STAGES (I_DIM / BK) // 16

typedef int v4i __attribute__((ext_vector_type(4)));
typedef __attribute__((address_space(1))) v4i* g_v4i_p; // global
typedef __attribute__((address_space(3))) v4i* l_v4i_p; // LDS

// ---- async global -> LDS copy of 16 bytes per lane --------------------------
#if __has_builtin(__builtin_amdgcn_global_load_async_to_lds_b128)
#define ASYNC_PATH 1
__device__ __forceinline__ void cp_async_b128(const float* gsrc, float* ldst) {
  __builtin_amdgcn_global_load_async_to_lds_b128((g_v4i_p)gsrc, (l_v4i_p)ldst,
                                                 0, 0);
}
#else
#define ASYNC_PATH 3
__device__ __forceinline__ void cp_async_b128(const float* gsrc, float* ldst) {
  unsigned loff =
      (unsigned)(uint64_t)(__attribute__((address_space(3))) void*)(void*)ldst;
  asm volatile("global_load_async_to_lds_b128 %0, %1, off" ::"v"(loff),
               "v"(gsrc)
               : "memory");
}
#endif

#if __has_builtin(__builtin_amdgcn_s_wait_asynccnt)
#define WAIT_ASYNC(n) __builtin_amdgcn_s_wait_asynccnt(n)
#else
#define WAIT_ASYNC(n) asm volatile("s_wait_asynccnt %0" ::"i"(n) : "memory")
#endif

union F4 {
  float4 v;
  float f[4];
};

__global__ __launch_bounds__(THREADS, 8) void minmax_mm_kernel(
    const float* __restrict__ x, const float* __restrict__ w,
    float* __restrict__ out) {
  // double-buffered tiles: xs[buf][row][k], ws[buf][k][col]
  __shared__ float xs[2][BM * BK]; // 2 * 8 KB
  __shared__ float ws[2][BK * BN]; // 2 * 8 KB

  const int tid = threadIdx.x;
  const int b0 = blockIdx.y * BM;
  const int o0 = blockIdx.x * BN;

  // copy mapping: x tile row = 32 floats = 128B -> 8 lanes of b128
  const int xr = tid >> 3;      // 0..31
  const int xc = (tid & 7) * 4; // float col in k
  // w tile row = 64 floats = 256B -> 16 lanes of b128
  const int wr = tid >> 4;       // 0..15
  const int wc = (tid & 15) * 4; // float col in o

  auto issue_stage = [&](int s, int buf) {
    const float* xg = x + (size_t)b0 * I_DIM + (size_t)s * BK;
    const float* wg = w + (size_t)(s * BK) * O_DIM + o0;
#pragma unroll
    for (int p = 0; p < 2; ++p) {
      const int r = xr + p * 32;
      cp_async_b128(xg + (size_t)r * I_DIM + xc, &xs[buf][r * BK + xc]);
    }
#pragma unroll
    for (int p = 0; p < 2; ++p) {
      const int r = wr + p * 16;
      cp_async_b128(wg + (size_t)r * O_DIM + wc, &ws[buf][r * BN + wc]);
    }
  };

  // compute mapping: 4x4 micro-tile per thread
  const int ty = tid >> 4; // 0..15 -> rows
  const int tx = tid & 15; // 0..15 -> cols
  const int r0 = ty * 4;
  const int c0 = tx * 4;

  float acc[4][4];
#pragma unroll
  for (int i = 0; i < 4; ++i)
#pragma unroll
    for (int j = 0; j < 4; ++j) acc[i][j] = __builtin_inff();

  issue_stage(0, 0);
  WAIT_ASYNC(0);
  __syncthreads();

  for (int s = 0; s < STAGES; ++s) {
    const int buf = s & 1;
    if (s + 1 < STAGES) issue_stage(s + 1, buf ^ 1);

    const float* xsb = &xs[buf][0];
    const float* wsb = &ws[buf][0];

    // Cap unroll at 2 k4-groups: keeps ~16 LDS loads in flight while holding
    // the live set (~100 VGPRs) under the 256-VGPR direct-address window.
#pragma unroll 2
    for (int k4 = 0; k4 < BK; k4 += 4) {
      F4 a[4];
#pragma unroll
      for (int i = 0; i < 4; ++i)
        a[i].v = *reinterpret_cast<const float4*>(&xsb[(r0 + i) * BK + k4]);
#pragma unroll
      for (int kk = 0; kk < 4; ++kk) {
        F4 b;
        b.v = *reinterpret_cast<const float4*>(&wsb[(k4 + kk) * BN + c0]);
#pragma unroll
        for (int i = 0; i < 4; ++i) {
          const float ai = a[i].f[kk];
#pragma unroll
          for (int j = 0; j < 4; ++j)
            acc[i][j] = fminf(acc[i][j], fmaxf(ai, b.f[j]));
        }
      }
    }

    if (s + 1 < STAGES) WAIT_ASYNC(0);
    __syncthreads();
  }

#pragma unroll
  for (int i = 0; i < 4; ++i) {
    float4 o;
    o.x = acc[i][0];
    o.y = acc[i][1];
    o.z = acc[i][2];
    o.w = acc[i][3];
    *reinterpret_cast<float4*>(out + (size_t)(b0 + r0 + i) * O_DIM +
                               (o0 + c0)) = o;
  }
}

extern "C" void kernel_launch(void* const* d_in, const int* in_sizes, int n_in,
                              void* d_out, int out_size, void* d_ws,
                              size_t ws_size, hipStream_t stream) {
  const float* x = (const float*)d_in[0]; // [B, I]
  const float* w = (const float*)d_in[1]; // [I, O]
  float* out = (float*)d_out;             // [B, O]
  (void)in_sizes;
  (void)n_in;
  (void)out_size;
  (void)d_ws;
  (void)ws_size;
  dim3 grid(O_DIM / BN, B_DIM / BM); // (8, 16)
  minmax_mm_kernel<<<grid, THREADS, 0, stream>>>(x, w, out);
}